// ModelWithEdgeFeatures_11467562680505
// MI455X (gfx1250) — compile-verified
//
#include <hip/hip_runtime.h>
#include <hip/hip_bf16.h>

// GNN message-passing stack, factorized:
//   agg[i] = (d(i)+1)*x[i] @ Wd^T + Sx[i] @ Ws^T + Sea[i] @ We^T + (d(i)+1)*b
// with Sx[i] = x[i] + sum_{e:dst=i} x[src_e],  Sea[i] = 1 + sum_{e:dst=i} ea_e.
// CSR built once -> atomic-free L2-resident gathers. Dense N x 144 @ 144 x 64
// GEMM on V_WMMA_F32_16X16X4_F32; the three concat sections are read in-place
// via three branch-free K-loops (keeps the WMMA stream software-pipelined).

typedef __attribute__((ext_vector_type(2))) float v2f;
typedef __attribute__((ext_vector_type(8))) float v8f;

#define CATDIM 144
#define HDIM   64
#define ECDIM  16

// ---------------------------------------------------------------- utility
__global__ void k_fill_f(float* __restrict__ p, float v, int n) {
  int i = blockIdx.x * blockDim.x + threadIdx.x;
  if (i < n) p[i] = v;
}
__global__ void k_zero_i(int* __restrict__ p, int n) {
  int i = blockIdx.x * blockDim.x + threadIdx.x;
  if (i < n) p[i] = 0;
}
__global__ void k_copy_i(const int* __restrict__ s, int* __restrict__ d, int n) {
  int i = blockIdx.x * blockDim.x + threadIdx.x;
  if (i < n) d[i] = s[i];
}

// ---------------------------------------------------------------- CSR build
__global__ void k_hist(const int* __restrict__ dst, int E, int* __restrict__ cnt) {
  int e = blockIdx.x * blockDim.x + threadIdx.x;
  if (e < E) atomicAdd(&cnt[dst[e]], 1);
}

// Single-block exclusive scan over N counts (N ~ 50000; ~200 tile iterations).
__global__ __launch_bounds__(256) void k_scan_rowptr(const int* __restrict__ cnt, int N,
                                                     int* __restrict__ rowptr) {
  __shared__ int tile[256];
  __shared__ int carry_s;
  if (threadIdx.x == 0) carry_s = 0;
  __syncthreads();
  for (int base = 0; base < N; base += 256) {
    int i = base + threadIdx.x;
    int v = (i < N) ? cnt[i] : 0;
    tile[threadIdx.x] = v;
    __syncthreads();
    for (int off = 1; off < 256; off <<= 1) {
      int t = (threadIdx.x >= off) ? tile[threadIdx.x - off] : 0;
      __syncthreads();
      tile[threadIdx.x] += t;
      __syncthreads();
    }
    int incl = tile[threadIdx.x];
    int carry = carry_s;                       // uniform read of previous carry
    if (i < N) rowptr[i] = carry + incl - v;   // exclusive
    __syncthreads();
    if (threadIdx.x == 255) carry_s = carry + tile[255];
    __syncthreads();
  }
  if (threadIdx.x == 0) rowptr[N] = carry_s;
}

__global__ void k_fill_csr(const int* __restrict__ src, const int* __restrict__ dst, int E,
                           int* __restrict__ cursor, int* __restrict__ csr_src,
                           int* __restrict__ csr_e) {
  int e = blockIdx.x * blockDim.x + threadIdx.x;
  if (e < E) {
    int pos = atomicAdd(&cursor[dst[e]], 1);
    csr_src[pos] = src[e];
    csr_e[pos]   = e;
  }
}

__global__ void k_degp1(const int* __restrict__ rowptr, int N, float* __restrict__ degp1) {
  int i = blockIdx.x * blockDim.x + threadIdx.x;
  if (i < N) degp1[i] = (float)(rowptr[i + 1] - rowptr[i] + 1);
}

// ---------------------------------------------------------------- aggregation (gather, no atomics)
__global__ __launch_bounds__(256) void k_aggr_ea(const int* __restrict__ rowptr,
                                                 const int* __restrict__ csr_e,
                                                 const float* __restrict__ ea, int N,
                                                 float* __restrict__ Sea) {
  int t = blockIdx.x * blockDim.x + threadIdx.x;
  int i = t >> 3, c2 = (t & 7) * 2;
  if (i < N) {
    float ax = 1.0f, ay = 1.0f;
    int e0 = rowptr[i], e1 = rowptr[i + 1];
    for (int j = e0; j < e1; ++j) {
      const float2 v = *(const float2*)(ea + (size_t)csr_e[j] * ECDIM + c2);
      ax += v.x; ay += v.y;
    }
    *(float2*)(Sea + (size_t)i * ECDIM + c2) = make_float2(ax, ay);
  }
}

__global__ __launch_bounds__(256) void k_aggr_x(const int* __restrict__ rowptr,
                                                const int* __restrict__ csr_src,
                                                const float* __restrict__ x, int N,
                                                float* __restrict__ Sx) {
  int t = blockIdx.x * blockDim.x + threadIdx.x;
  int i = t >> 5, c2 = (t & 31) * 2;
  if (i < N) {
    float2 self = *(const float2*)(x + (size_t)i * HDIM + c2);
    float ax = self.x, ay = self.y;
    int e0 = rowptr[i], e1 = rowptr[i + 1];
    for (int j = e0; j < e1; ++j) {
      const float2 v = *(const float2*)(x + (size_t)csr_src[j] * HDIM + c2);
      ax += v.x; ay += v.y;
    }
    *(float2*)(Sx + (size_t)i * HDIM + c2) = make_float2(ax, ay);
  }
}

// ---------------------------------------------------------------- WMMA GEMM
// out = relu( [ (d+1)*x | Sx | Sea ] @ W^T + (d+1)*bias ), concat read in-place
// via three straight-line K-sections (no branches inside any K-loop).
__global__ __launch_bounds__(256) void k_gemm_wmma(
    const float* __restrict__ x, const float* __restrict__ Sx,
    const float* __restrict__ Sea, const float* __restrict__ degp1,
    const float* __restrict__ W, const float* __restrict__ bias,
    float* __restrict__ out, int nrows) {
  __shared__ float sW[HDIM * CATDIM];
  __shared__ float sB[HDIM];
  for (int idx = threadIdx.x; idx < HDIM * CATDIM; idx += 256) sW[idx] = W[idx];
  if (threadIdx.x < HDIM) sB[threadIdx.x] = bias[threadIdx.x];
  __syncthreads();

  const int wave = threadIdx.x >> 5;
  const int lane = threadIdx.x & 31;
  const int tile = blockIdx.x * 8 + wave;          // wave-uniform
  const int ntiles = (nrows + 15) >> 4;
  if (tile >= ntiles) return;

  const int r0 = tile * 16;
  const int m = lane & 15;
  const int half = lane >> 4;

  int arow = r0 + m; if (arow >= nrows) arow = nrows - 1;   // stores guarded below
  const float dscale = degp1[arow];
  const float* __restrict__ xr  = x   + (size_t)arow * HDIM;
  const float* __restrict__ sxr = Sx  + (size_t)arow * HDIM;
  const float* __restrict__ ser = Sea + (size_t)arow * ECDIM;
  __builtin_prefetch(xr, 0, 1);
  __builtin_prefetch(sxr, 0, 1);

  const float* __restrict__ w0 = sW + (0  + m) * CATDIM;
  const float* __restrict__ w1 = sW + (16 + m) * CATDIM;
  const float* __restrict__ w2 = sW + (32 + m) * CATDIM;
  const float* __restrict__ w3 = sW + (48 + m) * CATDIM;

  v8f c0 = {}, c1 = {}, c2 = {}, c3 = {};

#define WMMA_STEP(AVEC, KOFF)                                                              \
  do {                                                                                     \
    v2f b0v; b0v.x = w0[(KOFF)]; b0v.y = w0[(KOFF) + 1];                                   \
    v2f b1v; b1v.x = w1[(KOFF)]; b1v.y = w1[(KOFF) + 1];                                   \
    v2f b2v; b2v.x = w2[(KOFF)]; b2v.y = w2[(KOFF) + 1];                                   \
    v2f b3v; b3v.x = w3[(KOFF)]; b3v.y = w3[(KOFF) + 1];                                   \
    c0 = __builtin_amdgcn_wmma_f32_16x16x4_f32(false, (AVEC), false, b0v, (short)0, c0,    \
                                               false, false);                              \
    c1 = __builtin_amdgcn_wmma_f32_16x16x4_f32(false, (AVEC), false, b1v, (short)0, c1,    \
                                               false, false);                              \
    c2 = __builtin_amdgcn_wmma_f32_16x16x4_f32(false, (AVEC), false, b2v, (short)0, c2,    \
                                               false, false);                              \
    c3 = __builtin_amdgcn_wmma_f32_16x16x4_f32(false, (AVEC), false, b3v, (short)0, c3,    \
                                               false, false);                              \
  } while (0)

  // Section 1: K = 0..63, A = (d+1) * x
  #pragma unroll 4
  for (int k = 0; k < HDIM; k += 4) {
    const int kk = k + 2 * half;
    v2f a; a.x = dscale * xr[kk]; a.y = dscale * xr[kk + 1];
    WMMA_STEP(a, kk);
  }
  // Section 2: K = 64..127, A = Sx
  #pragma unroll 4
  for (int k = 0; k < HDIM; k += 4) {
    const int kk = k + 2 * half;
    v2f a; a.x = sxr[kk]; a.y = sxr[kk + 1];
    WMMA_STEP(a, HDIM + kk);
  }
  // Section 3: K = 128..143, A = Sea
  #pragma unroll
  for (int k = 0; k < ECDIM; k += 4) {
    const int kk = k + 2 * half;
    v2f a; a.x = ser[kk]; a.y = ser[kk + 1];
    WMMA_STEP(a, 2 * HDIM + kk);
  }
#undef WMMA_STEP

  #pragma unroll
  for (int j = 0; j < 8; ++j) {
    const int row = r0 + j + 8 * half;
    if (row < nrows) {
      const float db = degp1[row];
      float v0 = fmaxf(c0[j] + db * sB[0  + m], 0.f);
      float v1 = fmaxf(c1[j] + db * sB[16 + m], 0.f);
      float v2 = fmaxf(c2[j] + db * sB[32 + m], 0.f);
      float v3 = fmaxf(c3[j] + db * sB[48 + m], 0.f);
      float* o = out + (size_t)row * HDIM;
      o[0 + m] = v0; o[16 + m] = v1; o[32 + m] = v2; o[48 + m] = v3;
    }
  }
}

// ---------------------------------------------------------------- batchnorm
__global__ __launch_bounds__(256) void k_bn_stats(const float* __restrict__ h, int N,
                                                  float* __restrict__ sums) {
  __shared__ float sdat[2][4][HDIM];
  const int c = threadIdx.x & 63;
  const int rl = threadIdx.x >> 6;
  const int base = blockIdx.x * 256;
  const int lim = (base + 256 < N) ? base + 256 : N;
  float s = 0.f, s2 = 0.f;
  for (int r = base + rl; r < lim; r += 4) {
    float v = h[(size_t)r * HDIM + c];
    s += v; s2 += v * v;
  }
  sdat[0][rl][c] = s; sdat[1][rl][c] = s2;
  __syncthreads();
  if (rl == 0) {
    float ts = sdat[0][0][c] + sdat[0][1][c] + sdat[0][2][c] + sdat[0][3][c];
    float t2 = sdat[1][0][c] + sdat[1][1][c] + sdat[1][2][c] + sdat[1][3][c];
    atomicAdd(&sums[c], ts);
    atomicAdd(&sums[HDIM + c], t2);
  }
}

__global__ void k_bn_final(const float* __restrict__ sums, const float* __restrict__ g,
                           const float* __restrict__ be, float invN, float* __restrict__ ss) {
  int c = threadIdx.x;
  if (c < HDIM) {
    float mu  = sums[c] * invN;
    float var = sums[HDIM + c] * invN - mu * mu;
    float sc  = g[c] * rsqrtf(var + 1e-5f);
    ss[c]        = sc;
    ss[HDIM + c] = be[c] - mu * sc;
  }
}

__global__ void k_bn_apply(const float* __restrict__ h, const float* __restrict__ ss,
                           float* __restrict__ xo, int n) {
  int t = blockIdx.x * blockDim.x + threadIdx.x;
  if (t < n) {
    int c = t & (HDIM - 1);
    xo[t] = fmaxf(ss[c] * h[t] + ss[HDIM + c], 0.f);
  }
}

// ---------------------------------------------------------------- driver
extern "C" void kernel_launch(void* const* d_in, const int* in_sizes, int n_in,
                              void* d_out, int out_size, void* d_ws, size_t ws_size,
                              hipStream_t stream) {
  const float* x    = (const float*)d_in[0];
  const int*   esrc = (const int*)d_in[1];
  const int*   edst = (const int*)d_in[2];
  const float* ea   = (const float*)d_in[3];
  const float* Wl[3] = { (const float*)d_in[4], (const float*)d_in[6], (const float*)d_in[8] };
  const float* bl[3] = { (const float*)d_in[5], (const float*)d_in[7], (const float*)d_in[9] };
  const float* gl[2]  = { (const float*)d_in[10], (const float*)d_in[12] };
  const float* bel[2] = { (const float*)d_in[11], (const float*)d_in[13] };

  const int N = in_sizes[0] / HDIM;
  const int E = in_sizes[1];

  char* wsb = (char*)d_ws;
  size_t off = 0;
  auto carve = [&](size_t nbytes) {
    void* p = wsb + off;
    off += (nbytes + 255) & ~(size_t)255;
    return p;
  };
  int* cnt     = (int*)carve((size_t)N * 4);          // reused as cursor
  int* rowptr  = (int*)carve((size_t)(N + 1) * 4);
  int* csr_src = (int*)carve((size_t)E * 4);
  int* csr_e   = (int*)carve((size_t)E * 4);
  float* degp1 = (float*)carve((size_t)N * 4);
  float* Sea   = (float*)carve((size_t)N * ECDIM * 4);
  float* Sx    = (float*)carve((size_t)N * HDIM * 4);
  float* h     = (float*)carve((size_t)N * HDIM * 4);
  float* xb    = (float*)carve((size_t)N * HDIM * 4);
  float* stats = (float*)carve(256 * 4);              // sums(128) | scale+shift(128)
  float* ss    = stats + 128;

  const int B = 256;
  auto cdiv = [](long long a, long long b) { return (int)((a + b - 1) / b); };

  // --- CSR build (once; reused by all 3 layers and the edge-attr pass)
  k_zero_i<<<cdiv(N, B), B, 0, stream>>>(cnt, N);
  k_hist<<<cdiv(E, B), B, 0, stream>>>(edst, E, cnt);
  k_scan_rowptr<<<1, 256, 0, stream>>>(cnt, N, rowptr);
  k_copy_i<<<cdiv(N, B), B, 0, stream>>>(rowptr, cnt, N);     // cnt -> cursor
  k_fill_csr<<<cdiv(E, B), B, 0, stream>>>(esrc, edst, E, cnt, csr_src, csr_e);
  k_degp1<<<cdiv(N, B), B, 0, stream>>>(rowptr, N, degp1);

  // --- Sea = 1 + aggregate(edge_attr), layer-invariant
  k_aggr_ea<<<cdiv((long long)N * 8, B), B, 0, stream>>>(rowptr, csr_e, ea, N, Sea);

  const int ntiles = (N + 15) >> 4;
  const int gemm_blocks = (ntiles + 7) >> 3;

  for (int layer = 0; layer < 3; ++layer) {
    const float* xin = (layer == 0) ? x : xb;
    float* gout = (layer == 2) ? (float*)d_out : h;

    // Sx = self + gather of neighbors (atomic-free, L2-resident)
    k_aggr_x<<<cdiv((long long)N * 32, B), B, 0, stream>>>(rowptr, csr_src, xin, N, Sx);

    // h = relu( [ (d+1)*x | Sx | Sea ] @ W^T + (d+1)*b )  — WMMA f32
    k_gemm_wmma<<<gemm_blocks, 256, 0, stream>>>(xin, Sx, Sea, degp1,
                                                 Wl[layer], bl[layer], gout, N);

    if (layer < 2) {
      k_fill_f<<<1, 128, 0, stream>>>(stats, 0.0f, 128);
      k_bn_stats<<<cdiv(N, 256), 256, 0, stream>>>(h, N, stats);
      k_bn_final<<<1, 64, 0, stream>>>(stats, gl[layer], bel[layer], 1.0f / (float)N, ss);
      k_bn_apply<<<cdiv((long long)N * HDIM, B), B, 0, stream>>>(h, ss, xb, N * HDIM);
    }
  }
}